// ALoRaAtt_35201551958312
// MI455X (gfx1250) — compile-verified
//
#include <hip/hip_runtime.h>
#include <hip/hip_bf16.h>

typedef __attribute__((ext_vector_type(2))) float v2f;
typedef __attribute__((ext_vector_type(4))) float v4f;
typedef __attribute__((ext_vector_type(8))) float v8f;

namespace {
constexpr int Bc = 2, Lc = 2048, Hc = 8, Ec = 64;
constexpr int NT  = Lc / 16;      // 128 column tiles per head row
constexpr int WPB = 8;            // waves per block
constexpr int RS  = Hc * Ec;      // row stride in floats (512)
constexpr int LDSS = 20;          // padded LDS row stride (floats) -> conflict-free
}

__global__ __launch_bounds__(WPB * 32) void attn_fwd_wmma(
    const float* __restrict__ Q, const float* __restrict__ K,
    const float* __restrict__ V, float* __restrict__ Vout,
    float* __restrict__ SA)
{
  __shared__ float ldsP[WPB][16 * LDSS];

  const int lane = threadIdx.x & 31;
  const int wave = threadIdx.x >> 5;
  const int t  = blockIdx.x * WPB + wave;   // global row-tile id, 0..2047
  const int i  = t % NT;                    // row tile within head
  const int bh = t / NT;
  const int h  = bh % Hc;
  const int b  = bh / Hc;
  const int l0 = i * 16;

  const int m  = lane & 15;                 // row/col index within 16
  const int kh = lane >> 4;                 // which k-half (A/B frag layout)

  // ---- preload Q tile as A-matrix fragments: 16 k-steps of 16x4 (fp32) ----
  const float* qrow = Q + (((size_t)b * Lc + (l0 + m)) * Hc + h) * Ec;
  v2f qa[16];
#pragma unroll
  for (int st = 0; st < 16; ++st)
    qa[st] = *(const v2f*)(qrow + st * 4 + 2 * kh);

  v8f oacc[4] = {};                 // O accumulators: 16 rows x 64 cols
  float rsum[8] = {};               // per-row exp sums (C-layout groups)
  float* myP = ldsP[wave];
  const float scale = 0.125f;       // 1/sqrt(64)

  for (int j = 0; j <= i; ++j) {
    const int s0 = j * 16;
    const float* krow = K + (((size_t)b * Lc + (s0 + m)) * Hc + h) * Ec;
    if (j < i) __builtin_prefetch(krow + (size_t)16 * RS, 0, 0);  // next K tile

    // ---- S = Q * K^T : chain 16 x v_wmma_f32_16x16x4_f32 over E=64 ----
    v8f s = {};
#pragma unroll
    for (int st = 0; st < 16; ++st) {
      v2f kb = *(const v2f*)(krow + st * 4 + 2 * kh);
      s = __builtin_amdgcn_wmma_f32_16x16x4_f32(false, qa[st], false, kb,
                                                (short)0, s, false, false);
    }

    // ---- P = exp(scale*S) with causal mask; row sums; stage + raw SA store ----
    float* sabase = SA + ((size_t)bh * Lc + l0) * Lc + s0;
#pragma unroll
    for (int r = 0; r < 8; ++r) {
      const int row = r + 8 * kh;                  // local row M (C layout)
      float p = __expf(scale * s[r]);
      if (j == i && m > row) p = 0.0f;             // mask s > l (s0 == l0 here)
      myP[row * LDSS + m] = p;                     // stage for layout transpose
      sabase[(size_t)row * Lc + m] = p;            // unnormalized; fixed in sweep
      float q = p;                                 // row-sum across 16 lanes
      q += __shfl_xor(q, 1, 32);
      q += __shfl_xor(q, 2, 32);
      q += __shfl_xor(q, 4, 32);
      q += __shfl_xor(q, 8, 32);
      rsum[r] += q;
    }
    asm volatile("s_wait_dscnt 0" ::: "memory");

    // ---- O += P(16x16) * V(16x64) : 16 more WMMAs ----
    const float* vbase = V + (((size_t)b * Lc + s0) * Hc + h) * Ec;
#pragma unroll
    for (int ss = 0; ss < 16; ss += 4) {
      v2f a = *(const v2f*)(myP + m * LDSS + ss + 2 * kh);   // A-layout read
      const float* v0 = vbase + (size_t)(ss + 2 * kh) * RS + m;
#pragma unroll
      for (int d = 0; d < 4; ++d) {
        v2f bv;
        bv.x = v0[d * 16];
        bv.y = v0[RS + d * 16];
        oacc[d] = __builtin_amdgcn_wmma_f32_16x16x4_f32(false, a, false, bv,
                                                        (short)0, oacc[d],
                                                        false, false);
      }
    }
  }

  // ---- normalize O and write V-output (B,L,H,E) ----
#pragma unroll
  for (int r = 0; r < 8; ++r) {
    const float inv = 1.0f / rsum[r];
#pragma unroll
    for (int d = 0; d < 4; ++d) oacc[d][r] *= inv;
  }
  float* orow = Vout + (((size_t)b * Lc + l0) * Hc + h) * Ec + m;
#pragma unroll
  for (int r = 0; r < 8; ++r)
#pragma unroll
    for (int d = 0; d < 4; ++d)
      orow[(size_t)(r + 8 * kh) * RS + d * 16] = oacc[d][r];

  // ---- normalization sweep over SA row-tile (L2-resident) + zero fill ----
  asm volatile("s_wait_storecnt 0" ::: "memory");
  const int nend = (i + 1) * 16;
  const v4f z = {0.f, 0.f, 0.f, 0.f};
#pragma unroll
  for (int mm = 0; mm < 16; ++mm) {
    const float inv = 1.0f / __shfl(rsum[mm & 7], (mm & 8) ? 16 : 0, 32);
    float* sarow = SA + ((size_t)bh * Lc + (l0 + mm)) * Lc;
    for (int s4 = lane * 4; s4 < nend; s4 += 128) {
      v4f v = *(v4f*)(sarow + s4);
      v *= inv;
      *(v4f*)(sarow + s4) = v;
    }
    for (int s4 = nend + lane * 4; s4 < Lc; s4 += 128)
      *(v4f*)(sarow + s4) = z;
  }
}

extern "C" void kernel_launch(void* const* d_in, const int* in_sizes, int n_in,
                              void* d_out, int out_size, void* d_ws, size_t ws_size,
                              hipStream_t stream) {
  const float* Q = (const float*)d_in[0];
  const float* K = (const float*)d_in[1];
  const float* V = (const float*)d_in[2];
  float* Vout = (float*)d_out;
  float* SA   = (float*)d_out + (size_t)Bc * Lc * Hc * Ec;  // V first, then SA

  const int total_tiles = Bc * Hc * NT;          // 2048 waves
  const int blocks = total_tiles / WPB;          // 256 blocks of 8 waves
  attn_fwd_wmma<<<blocks, WPB * 32, 0, stream>>>(Q, K, V, Vout, SA);
}